// Pnet_post_27178553049680
// MI455X (gfx1250) — compile-verified
//
#include <hip/hip_runtime.h>

// ---------------- problem constants (match reference) ----------------
#define B_       8
#define H_       640
#define W_       640
#define N_       (H_ * W_)        // 409600 candidates per batch (w*h grid)
#define BN_      ((size_t)B_ * (size_t)N_)
#define MAX_DET_ 128
#define THRESH_  0.6f
#define NMS_IOU_ 0.3f

#define BLK      1024             // 32 wave32 waves per workgroup
#define NTILES   (N_ / BLK)       // 400, exact
#define NWORDS   (N_ / 32)        // 12800 bitmap words (51.2 KB LDS)

// ---------------------------------------------------------------------
// CDNA5 async global->LDS copy (ASYNCcnt-tracked). Generic pointers to
// LDS on gfx12xx carry the wave-relative LDS byte offset in their low
// 32 bits, which is exactly what the VDST operand of
// global_load_async_to_lds_b32 wants. GVS addressing: saddr (SGPR pair)
// + 32-bit per-lane byte offset.
// ---------------------------------------------------------------------
__device__ __forceinline__ void async_ld_b32(float* lds_dst,
                                             const float* sbase,
                                             unsigned voff_bytes) {
  unsigned ldsa = (unsigned)(unsigned long long)lds_dst;
  asm volatile("global_load_async_to_lds_b32 %0, %1, %2"
               :
               : "v"(ldsa), "v"(voff_bytes), "s"(sbase)
               : "memory");
}
__device__ __forceinline__ void wait_async_le5() {
  asm volatile("s_wait_asynccnt 5" ::: "memory");
}
__device__ __forceinline__ void wait_async_0() {
  asm volatile("s_wait_asynccnt 0" ::: "memory");
}

// ---------------------------------------------------------------------
// Stage 1: grid-cell -> clipped square candidate box. Reads coalesced in
// i (inner W dim of cls/roi), transposes through LDS so the SoA writes
// (candidate order n = i*H + j, j fast) are coalesced too. Branchless:
// invalid cells (s < THRESH) still get a finite box; they are masked out
// once in the NMS kernel's bitmap, never read again.
// ---------------------------------------------------------------------
__global__ void __launch_bounds__(1024)
pnet_map_kernel(const float* __restrict__ cls, const float* __restrict__ roi,
                const float* __restrict__ scl,
                const float* __restrict__ hraw, const float* __restrict__ wraw,
                float* __restrict__ X1, float* __restrict__ Y1,
                float* __restrict__ X2, float* __restrict__ Y2,
                float* __restrict__ S) {
  __shared__ float t[5][32][33];  // +1 pad: conflict-free transpose

  const int b  = blockIdx.z;
  const int i0 = blockIdx.x * 32;
  const int j0 = blockIdx.y * 32;
  const int tx = threadIdx.x, ty = threadIdx.y;
  const int i = i0 + tx;          // W index (x)
  const int j = j0 + ty;          // H index (y)

  const size_t p = ((size_t)(b * H_ + j) * W_ + i);
  const float  s  = cls[p * 2 + 1];              // score = cls[:,:,1].T
  const float4 o4 = ((const float4*)roi)[p];     // offset = roi.T (per cell)

  const float sc     = 1.0f / scl[b];
  const float stride = (float)(1279.0 / 639.0);  // (2*640+11-12)/(640-1)

  const float bxi = stride * (float)i;
  const float byj = stride * (float)j;
  // _tffix == trunc toward zero (args here are >= 0)
  const float bb1x = truncf(bxi * sc);
  const float bb1y = truncf(byj * sc);
  const float bb2x = truncf((bxi + 11.0f) * sc);
  const float bb2y = truncf((byj + 11.0f) * sc);

  float x1 = bb1x + (o4.x * 12.0f) * sc;
  float y1 = bb1y + (o4.y * 12.0f) * sc;
  float x2 = bb2x + (o4.z * 12.0f) * sc;
  float y2 = bb2y + (o4.w * 12.0f) * sc;

  // rec2square
  const float w = x2 - x1, h = y2 - y1;
  const float l = fmaxf(w, h);
  const float nx1 = x1 + w * 0.5f - l * 0.5f;
  const float ny1 = y1 + h * 0.5f - l * 0.5f;
  const float nx2 = nx1 + l, ny2 = ny1 + l;

  // clip to raw image, swap if inverted
  const float cx1 = fmaxf(0.0f, nx1);
  const float cy1 = fmaxf(0.0f, ny1);
  const float cx2 = fminf(wraw[b], nx2);
  const float cy2 = fminf(hraw[b], ny2);
  const bool inv = (cx1 > cx2) || (cy1 > cy2);
  const float fx1 = inv ? fminf(cx1, cx2) : cx1;
  const float fy1 = inv ? fminf(cy1, cy2) : cy1;
  const float fx2 = inv ? fmaxf(cx1, cx2) : cx2;
  const float fy2 = inv ? fmaxf(cy1, cy2) : cy2;

  t[0][ty][tx] = fx1;  t[1][ty][tx] = fy1;
  t[2][ty][tx] = fx2;  t[3][ty][tx] = fy2;
  t[4][ty][tx] = s;
  __syncthreads();

  // transposed, coalesced write: candidate n = i*H + j with j fastest
  const size_t slot = (size_t)b * N_ + (size_t)(i0 + ty) * H_ + (j0 + tx);
  X1[slot] = t[0][tx][ty];
  Y1[slot] = t[1][tx][ty];
  X2[slot] = t[2][tx][ty];
  Y2[slot] = t[3][tx][ty];
  S [slot] = t[4][tx][ty];
}

// ---------------------------------------------------------------------
// Stage 2: greedy NMS, one workgroup (32 waves) per batch.
// Iteration k fuses "suppress vs winner[k-1]" with "argmax for winner[k]"
// in a single sweep over the candidate stream, which is double-buffered
// into LDS via async-to-LDS loads. Suppression state = LDS bitmap
// (ds_or), argmax = unique 64-bit keys reduced with ds_max_u64.
// key = score_bits<<32 | (N-1-n)  -> max key == reference argmax with
// first-index tie-break. key==0 is the "nothing alive" sentinel (valid
// scores >= 0.6 give keys >= 0x3F19999A_00000000).
// ---------------------------------------------------------------------
__global__ void __launch_bounds__(BLK, 1)
pnet_nms_kernel(const float* __restrict__ X1, const float* __restrict__ Y1,
                const float* __restrict__ X2, const float* __restrict__ Y2,
                const float* __restrict__ S, float* __restrict__ out) {
  __shared__ unsigned           supp[NWORDS];     // 51.2 KB
  __shared__ float              tile[2][5][BLK];  // 40 KB double buffer
  __shared__ unsigned long long s_key;
  __shared__ float              s_box[5];

  const int b   = blockIdx.x;
  const int tid = threadIdx.x;
  const float* x1p = X1 + (size_t)b * N_;
  const float* y1p = Y1 + (size_t)b * N_;
  const float* x2p = X2 + (size_t)b * N_;
  const float* y2p = Y2 + (size_t)b * N_;
  const float* sp  = S  + (size_t)b * N_;

  // bitmap init: bit=1 means "never pickable" (below threshold)
  for (int w = tid; w < NWORDS; w += BLK) {
    const float* q = sp + (size_t)w * 32;
    unsigned m = 0u;
#pragma unroll
    for (int e = 0; e < 32; ++e) m |= (q[e] < THRESH_) ? (1u << e) : 0u;
    supp[w] = m;
  }
  __syncthreads();

  bool  have_w = false;
  float wx1 = 0.f, wy1 = 0.f, wx2 = 0.f, wy2 = 0.f, warea = 1.f;

  for (int k = 0; k < MAX_DET_; ++k) {
    if (tid == 0) s_key = 0ull;
    __syncthreads();

    unsigned long long best = 0ull;

    // software pipeline: prefetch tile 0
    {
      const unsigned voff = (unsigned)tid * 4u;
      async_ld_b32(&tile[0][0][tid], x1p, voff);
      async_ld_b32(&tile[0][1][tid], y1p, voff);
      async_ld_b32(&tile[0][2][tid], x2p, voff);
      async_ld_b32(&tile[0][3][tid], y2p, voff);
      async_ld_b32(&tile[0][4][tid], sp,  voff);
    }

    for (int T = 0; T < NTILES; ++T) {
      const int cur = T & 1;
      if (T + 1 < NTILES) {
        const int      nb   = (T + 1) & 1;
        const unsigned voff = (unsigned)((T + 1) * BLK + tid) * 4u;
        async_ld_b32(&tile[nb][0][tid], x1p, voff);
        async_ld_b32(&tile[nb][1][tid], y1p, voff);
        async_ld_b32(&tile[nb][2][tid], x2p, voff);
        async_ld_b32(&tile[nb][3][tid], y2p, voff);
        async_ld_b32(&tile[nb][4][tid], sp,  voff);
        wait_async_le5();   // tile T's 5 loads done (in-order completion)
      } else {
        wait_async_0();
      }

      const int idx = T * BLK + tid;
      const unsigned wrd = supp[idx >> 5];
      if (!((wrd >> (idx & 31)) & 1u)) {
        const float x1 = tile[cur][0][tid];
        const float y1 = tile[cur][1][tid];
        const float x2 = tile[cur][2][tid];
        const float y2 = tile[cur][3][tid];
        const float sv = tile[cur][4][tid];
        bool kill = false;
        if (have_w) {
          const float xx1 = fmaxf(wx1, x1), yy1 = fmaxf(wy1, y1);
          const float xx2 = fminf(wx2, x2), yy2 = fminf(wy2, y2);
          const float inter =
              fmaxf(0.0f, xx2 - xx1 + 1.0f) * fmaxf(0.0f, yy2 - yy1 + 1.0f);
          const float area = (x2 - x1 + 1.0f) * (y2 - y1 + 1.0f);
          const float iou  = inter / (warea + area - inter);
          kill = iou > NMS_IOU_;
        }
        if (kill) {
          atomicOr(&supp[idx >> 5], 1u << (idx & 31));  // ds_or_b32
        } else {
          const unsigned long long key =
              ((unsigned long long)__float_as_uint(sv) << 32) |
              (unsigned)(N_ - 1 - idx);
          if (key > best) best = key;
        }
      }
    }

    atomicMax(&s_key, best);  // ds_max_u64: one-step block argmax
    __syncthreads();
    const unsigned long long key = s_key;

    if (key == 0ull) {  // nothing alive: remaining rows are zeros
      for (int kk = k; kk < MAX_DET_; ++kk)
        if (tid < 5) out[((size_t)b * MAX_DET_ + kk) * 5 + tid] = 0.0f;
      break;            // key is uniform -> uniform break
    }

    if (tid == 0) {
      const int slot = N_ - 1 - (int)(unsigned)(key & 0xFFFFFFFFu);
      const float x1 = x1p[slot], y1 = y1p[slot];
      const float x2 = x2p[slot], y2 = y2p[slot];
      s_box[0] = x1; s_box[1] = y1; s_box[2] = x2; s_box[3] = y2;
      s_box[4] = (x2 - x1 + 1.0f) * (y2 - y1 + 1.0f);
      const size_t o = ((size_t)b * MAX_DET_ + k) * 5;
      out[o + 0] = x1; out[o + 1] = y1;
      out[o + 2] = x2; out[o + 3] = y2;
      out[o + 4] = __uint_as_float((unsigned)(key >> 32));
    }
    __syncthreads();
    wx1 = s_box[0]; wy1 = s_box[1]; wx2 = s_box[2]; wy2 = s_box[3];
    warea = s_box[4];
    have_w = true;   // winner[k] suppresses itself next sweep (IoU==1)
  }
}

// ---------------------------------------------------------------------
extern "C" void kernel_launch(void* const* d_in, const int* in_sizes, int n_in,
                              void* d_out, int out_size, void* d_ws,
                              size_t ws_size, hipStream_t stream) {
  (void)in_sizes; (void)n_in; (void)out_size; (void)ws_size;
  const float* cls  = (const float*)d_in[0];  // (B,H,W,2)
  const float* roi  = (const float*)d_in[1];  // (B,H,W,4)
  const float* scl  = (const float*)d_in[2];  // (B,)
  const float* hraw = (const float*)d_in[3];  // (B,1)
  const float* wraw = (const float*)d_in[4];  // (B,1)
  float* out = (float*)d_out;                 // (B,128,5)

  // workspace: 5 SoA candidate arrays, 5 * 8 * 409600 * 4 = 65.5 MB
  float* X1 = (float*)d_ws;
  float* Y1 = X1 + BN_;
  float* X2 = Y1 + BN_;
  float* Y2 = X2 + BN_;
  float* S  = Y2 + BN_;

  pnet_map_kernel<<<dim3(W_ / 32, H_ / 32, B_), dim3(32, 32, 1), 0, stream>>>(
      cls, roi, scl, hraw, wraw, X1, Y1, X2, Y2, S);
  pnet_nms_kernel<<<dim3(B_), dim3(BLK), 0, stream>>>(X1, Y1, X2, Y2, S, out);
}